// ramsey_MPNN_66640712564859
// MI455X (gfx1250) — compile-verified
//
#include <hip/hip_runtime.h>
#include <hip/hip_bf16.h>
#include <math.h>

typedef __attribute__((ext_vector_type(2))) float v2f;
typedef __attribute__((ext_vector_type(8))) float v8f;

#define MODE_NONE    0
#define MODE_RELU    1
#define MODE_LEAKY   2
#define MODE_BN_RELU 3

// 1/sqrt(1 + 1e-5) : BatchNorm1d eval with running stats (0,1)
#define BN_INV_STD 0.99999500003749963f

// ---------------------------------------------------------------------------
// Column-wise inclusive prefix sum over rows (M=1024), split into 32 chunks.
// ---------------------------------------------------------------------------
__global__ void scan_partial_kernel(const float* __restrict__ in,
                                    float* __restrict__ part, int C) {
    int col   = blockIdx.x * blockDim.x + threadIdx.x;
    int chunk = blockIdx.y;               // 0..31, 32 rows each
    if (col >= C) return;
    int r0 = chunk * 32;
    float s = 0.f;
    #pragma unroll 4
    for (int r = 0; r < 32; ++r) s += in[(r0 + r) * C + col];
    part[chunk * C + col] = s;
}

__global__ void scan_final_kernel(const float* __restrict__ in,
                                  const float* __restrict__ part,
                                  float* __restrict__ out, int C) {
    int col   = blockIdx.x * blockDim.x + threadIdx.x;
    int chunk = blockIdx.y;
    if (col >= C) return;
    float run = 0.f;
    for (int c = 0; c < chunk; ++c) run += part[c * C + col];
    int r0 = chunk * 32;
    for (int r = 0; r < 32; ++r) {
        run += in[(r0 + r) * C + col];
        out[(r0 + r) * C + col] = run;
    }
}

// ---------------------------------------------------------------------------
// FP32 WMMA GEMM, fully shape-specialized: out = epilogue(A[1024,K]@W[K,N]+b).
// One wave per 16x16 C tile, V_WMMA_F32_16X16X4_F32 over the K loop.
// A frag (16x4): lanes 0-15 -> M=lane, K={k,k+1}; lanes 16-31 -> K={k+2,k+3}
// B frag (4x16): lanes 0-15 -> N=lane, K={k,k+1}; lanes 16-31 -> K={k+2,k+3}
// C/D (16x16): VGPR r, lane<16 -> M=r, lane>=16 -> M=r+8; N=lane&15
// Grid is exact (no guard), so EXEC is all-1s for every WMMA.
// ---------------------------------------------------------------------------
template <int K, int N, int MODE>
__global__ __launch_bounds__(128)
void gemm_wmma_kernel(const float* __restrict__ A, const float* __restrict__ W,
                      const float* __restrict__ bias,
                      const float* __restrict__ gamma,
                      const float* __restrict__ beta,
                      float* __restrict__ out) {
    constexpr int TILES_N = N / 16;
    const int wave = threadIdx.x >> 5;
    const int lane = threadIdx.x & 31;
    const int tile = blockIdx.x * 4 + wave;
    const int row0 = (tile / TILES_N) * 16;
    const int col0 = (tile % TILES_N) * 16;
    const int half = lane >> 4;              // 0: K pair {0,1}; 1: {2,3}
    const int sub  = lane & 15;

    // Base pointers absorb all per-lane terms; loop offsets are constants.
    const float* __restrict__ Arow = A + (row0 + sub) * K + 2 * half;
    const float* __restrict__ Wcol = W + 2 * half * N + col0 + sub;

    v8f acc = {};
    #pragma unroll 8
    for (int k = 0; k < K; k += 4) {
        v2f a, b;
        a.x = Arow[k];
        a.y = Arow[k + 1];
        b.x = Wcol[k * N];
        b.y = Wcol[k * N + N];
        acc = __builtin_amdgcn_wmma_f32_16x16x4_f32(
            false, a, false, b, (short)0, acc, false, false);
    }

    const int col = col0 + sub;
    const float bcol = bias ? bias[col] : 0.f;
    float gcol = 1.f, btcol = 0.f;
    if constexpr (MODE == MODE_BN_RELU) { gcol = gamma[col]; btcol = beta[col]; }

    #pragma unroll
    for (int r = 0; r < 8; ++r) {
        const int row = row0 + r + 8 * half;
        float v = acc[r] + bcol;
        if constexpr (MODE == MODE_BN_RELU) {
            v = gcol * v * BN_INV_STD + btcol;
            v = v > 0.f ? v : 0.f;
        } else if constexpr (MODE == MODE_RELU) {
            v = v > 0.f ? v : 0.f;
        } else if constexpr (MODE == MODE_LEAKY) {
            v = v >= 0.f ? v : 0.01f * v;
        }
        out[row * N + col] = v;
    }
}

// ---------------------------------------------------------------------------
// Edge predictor: for every pair s<d,
//   p = sigmoid( sum_k relu(Ap[s,k] + Bp[d,k]) * w2[k] + b2 )
// written to out[s,d] and out[d,s]; diagonal zeroed.
// Tile: 16 s-rows x 32 d-rows per 512-thread block (wave shares s, varies d).
// Bp staged in LDS with stride 257 -> bank (d*257+k)%64 distinct across d.
// ---------------------------------------------------------------------------
__global__ __launch_bounds__(512)
void edge_kernel(const float* __restrict__ Ap, const float* __restrict__ Bp,
                 const float* __restrict__ w2, const float* __restrict__ b2,
                 float* __restrict__ out, int n) {
    const int bj = blockIdx.x;   // d tile: 32 rows
    const int bi = blockIdx.y;   // s tile: 16 rows
    if (bi * 16 > bj * 32 + 31) return;   // whole tile is strictly lower-tri

    __shared__ float sA[16][256];
    __shared__ float sB[32][257];
    __shared__ float sw[256];

    const int tid = threadIdx.x;  // 0..511
    for (int i = tid; i < 16 * 256; i += 512) {
        sA[i >> 8][i & 255] = Ap[(size_t)(bi * 16 + (i >> 8)) * 256 + (i & 255)];
    }
    for (int i = tid; i < 32 * 256; i += 512) {
        sB[i >> 8][i & 255] = Bp[(size_t)(bj * 32 + (i >> 8)) * 256 + (i & 255)];
    }
    if (tid < 256) sw[tid] = w2[tid];
    __syncthreads();

    const int si = tid >> 5;     // 0..15
    const int di = tid & 31;     // 0..31
    const int s = bi * 16 + si;
    const int d = bj * 32 + di;
    if (s > d) return;
    if (s == d) { out[(size_t)s * n + d] = 0.f; return; }

    const float* __restrict__ ar = sA[si];
    const float* __restrict__ br = sB[di];
    float acc = 0.f;
    #pragma unroll 8
    for (int k = 0; k < 256; ++k) {
        float h = ar[k] + br[k];
        h = h > 0.f ? h : 0.f;
        acc = fmaf(h, sw[k], acc);
    }
    const float y = acc + b2[0];
    const float p = 1.f / (1.f + __expf(-y));
    out[(size_t)s * n + d] = p;
    out[(size_t)d * n + s] = p;
}

// ---------------------------------------------------------------------------
extern "C" void kernel_launch(void* const* d_in, const int* in_sizes, int n_in,
                              void* d_out, int out_size, void* d_ws, size_t ws_size,
                              hipStream_t stream) {
    const float* x   = (const float*)d_in[0];
    const float* w1a = (const float*)d_in[1];
    const float* b1a = (const float*)d_in[2];
    const float* g1  = (const float*)d_in[3];
    const float* bt1 = (const float*)d_in[4];
    const float* w1b = (const float*)d_in[5];
    const float* b1b = (const float*)d_in[6];
    const float* w2a = (const float*)d_in[7];
    const float* b2a = (const float*)d_in[8];
    const float* g2  = (const float*)d_in[9];
    const float* bt2 = (const float*)d_in[10];
    const float* w2b = (const float*)d_in[11];
    const float* b2b = (const float*)d_in[12];
    const float* lw1 = (const float*)d_in[13];
    const float* lb1 = (const float*)d_in[14];
    const float* lw2 = (const float*)d_in[15];
    const float* lb2 = (const float*)d_in[16];
    const float* ew1 = (const float*)d_in[17];
    const float* eb1 = (const float*)d_in[18];
    const float* ew2 = (const float*)d_in[19];
    const float* eb2 = (const float*)d_in[20];
    float* out = (float*)d_out;

    const int M = 1024, F = 128, H = 256;

    char* ws = (char*)d_ws;
    float* buf0 = (float*)(ws);                         // 1 MB slots
    float* buf1 = (float*)(ws + (1u << 20));
    float* buf2 = (float*)(ws + (2u << 20));
    float* part = (float*)(ws + (3u << 20));            // 32x256 partials

    // blocks = (M/16 * N/16) / 4 waves-per-block; exact, no tail.
    const int blkMH = (M / 16) * (H / 16) / 4;          // N=256 -> 256 blocks
    const int blkMF = (M / 16) * (F / 16) / 4;          // N=128 -> 128 blocks

    // ---- GIN layer 1: S1 = cumsum(x); T1 = BN+ReLU(S1@w1a+b1a); X1 = ReLU(T1@w1b+b1b)
    scan_partial_kernel<<<dim3(F / 64, 32), 64, 0, stream>>>(x, part, F);
    scan_final_kernel<<<dim3(F / 64, 32), 64, 0, stream>>>(x, part, buf0, F);
    gemm_wmma_kernel<128, 256, MODE_BN_RELU><<<blkMH, 128, 0, stream>>>(
        buf0, w1a, b1a, g1, bt1, buf1);
    gemm_wmma_kernel<256, 256, MODE_RELU><<<blkMH, 128, 0, stream>>>(
        buf1, w1b, b1b, nullptr, nullptr, buf2);
    // (leaky after ReLU output is the identity)

    // ---- GIN layer 2
    scan_partial_kernel<<<dim3(H / 64, 32), 64, 0, stream>>>(buf2, part, H);
    scan_final_kernel<<<dim3(H / 64, 32), 64, 0, stream>>>(buf2, part, buf0, H);
    gemm_wmma_kernel<256, 256, MODE_BN_RELU><<<blkMH, 128, 0, stream>>>(
        buf0, w2a, b2a, g2, bt2, buf1);
    gemm_wmma_kernel<256, 128, MODE_RELU><<<blkMF, 128, 0, stream>>>(
        buf1, w2b, b2b, nullptr, nullptr, buf2);

    // ---- node MLP head: T5 = leaky(X2@lw1+lb1); X3 = T5@lw2+lb2
    gemm_wmma_kernel<128, 256, MODE_LEAKY><<<blkMH, 128, 0, stream>>>(
        buf2, lw1, lb1, nullptr, nullptr, buf0);
    gemm_wmma_kernel<256, 128, MODE_NONE><<<blkMF, 128, 0, stream>>>(
        buf0, lw2, lb2, nullptr, nullptr, buf1);

    // ---- edge-MLP decomposition: Ap = X3@ew1[:F] + eb1 ; Bp = X3@ew1[F:]
    gemm_wmma_kernel<128, 256, MODE_NONE><<<blkMH, 128, 0, stream>>>(
        buf1, ew1, eb1, nullptr, nullptr, buf2);
    gemm_wmma_kernel<128, 256, MODE_NONE><<<blkMH, 128, 0, stream>>>(
        buf1, ew1 + (size_t)F * H, nullptr, nullptr, nullptr, buf0);

    // ---- fused relu-dot + sigmoid over all pairs, symmetric scatter
    edge_kernel<<<dim3(32, 64), 512, 0, stream>>>(buf2, buf0, ew2, eb2, out, M);
}